// SAGE_delta_77824807403623
// MI455X (gfx1250) — compile-verified
//
#include <hip/hip_runtime.h>

typedef __attribute__((ext_vector_type(2))) float v2f;
typedef __attribute__((ext_vector_type(8))) float v8f;

// ---------------------------------------------------------------------------
// degree count: atomic +1.0f per edge into deg[dst]
// ---------------------------------------------------------------------------
__global__ void sage_degree_kernel(const int* __restrict__ dst,
                                   float* __restrict__ deg, int E) {
  int e = blockIdx.x * blockDim.x + threadIdx.x;
  if (e < E) atomicAdd(&deg[dst[e]], 1.0f);
}

// deg -> 1/max(deg,1), in place
__global__ void sage_dinv_kernel(float* __restrict__ deg, int n) {
  int i = blockIdx.x * blockDim.x + threadIdx.x;
  if (i < n) {
    float d = deg[i];
    deg[i] = 1.0f / fmaxf(d, 1.0f);
  }
}

// flags: keep -> 1, low -> 2 (sets are disjoint)
__global__ void sage_flags_kernel(const int* __restrict__ keep, int nk,
                                  const int* __restrict__ low, int nl,
                                  int* __restrict__ flags) {
  int i = blockIdx.x * blockDim.x + threadIdx.x;
  if (i < nk) flags[keep[i]] = 1;
  if (i < nl) flags[low[i]] = 2;
}

// ---------------------------------------------------------------------------
// edge aggregation: one wave32 per edge, lane moves float4 (128 floats/edge)
// agg[dst] += x[src]   (deg_inv applied later in the GEMM)
// ---------------------------------------------------------------------------
__global__ void sage_aggregate_kernel(const float* __restrict__ x,
                                      const int* __restrict__ src,
                                      const int* __restrict__ dst,
                                      float* __restrict__ agg, int E) {
  const int lane = threadIdx.x & 31;
  const int wave = threadIdx.x >> 5;
  const int e = blockIdx.x * 8 + wave;
  if (e >= E) return;
  const int s = src[e];
  const int d = dst[e];
  const float4 v = ((const float4*)(x + (size_t)s * 128))[lane];
  float* ad = agg + (size_t)d * 128 + lane * 4;
  atomicAdd(ad + 0, v.x);
  atomicAdd(ad + 1, v.y);
  atomicAdd(ad + 2, v.z);
  atomicAdd(ad + 3, v.w);
}

// ---------------------------------------------------------------------------
// Layer 1: h1 = relu(x @ Wself + (agg*dinv) @ Wneigh + b)      [N x 128]
// block = 256 (8 waves); wave w -> 16-col tile w; blockIdx -> 16-row stripe.
// f32 WMMA 16x16x4, K marched in steps of 4, both products share one
// accumulator (D = A*B + C chained).
// ---------------------------------------------------------------------------
__global__ void __launch_bounds__(256)
sage_gemm1_kernel(const float* __restrict__ x, const float* __restrict__ agg,
                  const float* __restrict__ dinv,
                  const float* __restrict__ Wself,
                  const float* __restrict__ Wneigh,
                  const float* __restrict__ bias,
                  float* __restrict__ out, int n_rows) {
  const int lane = threadIdx.x & 31;
  const int wave = threadIdx.x >> 5;      // 0..7 -> column tile
  const int row0 = blockIdx.x * 16;
  const int col0 = wave * 16;
  const int m  = lane & 15;               // A row / B col / C col index
  const int kh = lane >> 4;               // K half (0/1)

  int rowA = row0 + m;
  int rowC = rowA < n_rows ? rowA : (n_rows - 1);   // clamp: keep EXEC full
  const float di = dinv[rowC];
  const float* xrow = x   + (size_t)rowC * 128;
  const float* arow = agg + (size_t)rowC * 128;

  v8f c = {};
#pragma unroll
  for (int k0 = 0; k0 < 128; k0 += 4) {
    const int k = k0 + 2 * kh;
    v2f a_s, a_n, b_s, b_n;
    a_s.x = xrow[k];
    a_s.y = xrow[k + 1];
    a_n.x = arow[k] * di;
    a_n.y = arow[k + 1] * di;
    b_s.x = Wself[(size_t)k * 128 + col0 + m];
    b_s.y = Wself[(size_t)(k + 1) * 128 + col0 + m];
    b_n.x = Wneigh[(size_t)k * 128 + col0 + m];
    b_n.y = Wneigh[(size_t)(k + 1) * 128 + col0 + m];
    c = __builtin_amdgcn_wmma_f32_16x16x4_f32(false, a_s, false, b_s,
                                              (short)0, c, false, false);
    c = __builtin_amdgcn_wmma_f32_16x16x4_f32(false, a_n, false, b_n,
                                              (short)0, c, false, false);
  }

  const float bv = bias[col0 + m];
  const int mrow0 = row0 + 8 * kh;        // C/D layout: lanes 16-31 hold M=8..15
#pragma unroll
  for (int r = 0; r < 8; ++r) {
    const int row = mrow0 + r;
    if (row < n_rows) {
      float v = c[r] + bv;
      out[(size_t)row * 128 + col0 + m] = v > 0.0f ? v : 0.0f;
    }
  }
}

// ---------------------------------------------------------------------------
// Layer 2 + output fixup:
//   h2 = h1 @ W2self + (agg*dinv) @ W2neigh + b2          [N x 64]
//   out = keep ? embedding : (low ? 2*h2 : h2)
// block = 128 (4 waves covering 64 cols)
// ---------------------------------------------------------------------------
__global__ void __launch_bounds__(128)
sage_gemm2_kernel(const float* __restrict__ h1, const float* __restrict__ agg,
                  const float* __restrict__ dinv,
                  const float* __restrict__ Wself,
                  const float* __restrict__ Wneigh,
                  const float* __restrict__ bias,
                  const float* __restrict__ embedding,
                  const int* __restrict__ flags,
                  float* __restrict__ out, int n_rows) {
  const int lane = threadIdx.x & 31;
  const int wave = threadIdx.x >> 5;      // 0..3 -> column tile
  const int row0 = blockIdx.x * 16;
  const int col0 = wave * 16;
  const int m  = lane & 15;
  const int kh = lane >> 4;

  int rowA = row0 + m;
  int rowC = rowA < n_rows ? rowA : (n_rows - 1);
  const float di = dinv[rowC];
  const float* xrow = h1  + (size_t)rowC * 128;
  const float* arow = agg + (size_t)rowC * 128;

  v8f c = {};
#pragma unroll
  for (int k0 = 0; k0 < 128; k0 += 4) {
    const int k = k0 + 2 * kh;
    v2f a_s, a_n, b_s, b_n;
    a_s.x = xrow[k];
    a_s.y = xrow[k + 1];
    a_n.x = arow[k] * di;
    a_n.y = arow[k + 1] * di;
    b_s.x = Wself[(size_t)k * 64 + col0 + m];
    b_s.y = Wself[(size_t)(k + 1) * 64 + col0 + m];
    b_n.x = Wneigh[(size_t)k * 64 + col0 + m];
    b_n.y = Wneigh[(size_t)(k + 1) * 64 + col0 + m];
    c = __builtin_amdgcn_wmma_f32_16x16x4_f32(false, a_s, false, b_s,
                                              (short)0, c, false, false);
    c = __builtin_amdgcn_wmma_f32_16x16x4_f32(false, a_n, false, b_n,
                                              (short)0, c, false, false);
  }

  const float bv = bias[col0 + m];
  const int mrow0 = row0 + 8 * kh;
#pragma unroll
  for (int r = 0; r < 8; ++r) {
    const int row = mrow0 + r;
    if (row < n_rows) {
      const int f = flags[row];
      float v = c[r] + bv;
      if (f == 2) v = 2.0f * v;
      if (f == 1) v = embedding[(size_t)row * 64 + col0 + m];
      out[(size_t)row * 64 + col0 + m] = v;
    }
  }
}

// ---------------------------------------------------------------------------
// launch
// ---------------------------------------------------------------------------
extern "C" void kernel_launch(void* const* d_in, const int* in_sizes, int n_in,
                              void* d_out, int out_size, void* d_ws, size_t ws_size,
                              hipStream_t stream) {
  const float* features  = (const float*)d_in[0];   // N x 128
  const float* W1_self   = (const float*)d_in[1];   // 128 x 128
  const float* W1_neigh  = (const float*)d_in[2];   // 128 x 128
  const float* b1        = (const float*)d_in[3];   // 128
  const float* W2_self   = (const float*)d_in[4];   // 128 x 64
  const float* W2_neigh  = (const float*)d_in[5];   // 128 x 64
  const float* b2        = (const float*)d_in[6];   // 64
  const float* embedding = (const float*)d_in[7];   // N x 64
  const int*   src       = (const int*)d_in[8];     // E
  const int*   dst       = (const int*)d_in[9];     // E
  const int*   keep_idx  = (const int*)d_in[10];
  const int*   low_idx   = (const int*)d_in[11];

  const int N  = in_sizes[0] / 128;
  const int E  = in_sizes[8];
  const int nk = in_sizes[10];
  const int nl = in_sizes[11];

  // workspace carve-up: dinv[N] | flags[N] | agg[N*128] (reused) | h1[N*128]
  float* ws    = (float*)d_ws;
  float* dinv  = ws;
  int*   flags = (int*)(ws + N);
  float* agg   = ws + 2 * (size_t)N;
  float* h1    = agg + (size_t)N * 128;

  hipMemsetAsync(dinv, 0, (size_t)N * sizeof(float), stream);
  hipMemsetAsync(flags, 0, (size_t)N * sizeof(int), stream);
  hipMemsetAsync(agg, 0, (size_t)N * 128 * sizeof(float), stream);

  sage_degree_kernel<<<(E + 255) / 256, 256, 0, stream>>>(dst, dinv, E);
  sage_dinv_kernel<<<(N + 255) / 256, 256, 0, stream>>>(dinv, N);
  const int nf = nk > nl ? nk : nl;
  sage_flags_kernel<<<(nf + 255) / 256, 256, 0, stream>>>(keep_idx, nk, low_idx, nl, flags);

  const int row_blocks = (N + 15) / 16;

  // layer 1
  sage_aggregate_kernel<<<(E + 7) / 8, 256, 0, stream>>>(features, src, dst, agg, E);
  sage_gemm1_kernel<<<row_blocks, 256, 0, stream>>>(features, agg, dinv,
                                                    W1_self, W1_neigh, b1, h1, N);

  // layer 2 (reuse agg)
  hipMemsetAsync(agg, 0, (size_t)N * 128 * sizeof(float), stream);
  sage_aggregate_kernel<<<(E + 7) / 8, 256, 0, stream>>>(h1, src, dst, agg, E);
  sage_gemm2_kernel<<<row_blocks, 128, 0, stream>>>(h1, agg, dinv,
                                                    W2_self, W2_neigh, b2,
                                                    embedding, flags,
                                                    (float*)d_out, N);
}